// Sampling_26972394619348
// MI455X (gfx1250) — compile-verified
//
#include <hip/hip_runtime.h>

typedef float v4f __attribute__((ext_vector_type(4)));

#define BATCH     262144
#define NSAMP     100
#define DIMS      3
#define W_FLOATS  (BATCH * NSAMP * DIMS)   // 78,643,200 floats of w
#define W_VECS    (W_FLOATS / 4)           // 19,660,800 float4s (75 per batch elem)
#define W_TILES   (W_VECS / 256)           // 76,800 tiles (exact)
#define L_FLOATS  (BATCH * NSAMP)          // 26,214,400 label floats
#define L_VECS    (L_FLOATS / 4)           // 6,553,600 float4s (25 per batch elem)

// w[e] = z_mean[b,d] + exp(0.5*z_log_var[b,d]) * eps[e]
// Streams epsilon through LDS with double-buffered async b128 loads (ASYNCcnt),
// computes 4 FMAs per vec, writes with non-temporal b128 stores.
__global__ __launch_bounds__(256) void sampling_w_kernel(
    const float* __restrict__ zm, const float* __restrict__ zlv,
    const float* __restrict__ eps, float* __restrict__ out)
{
    // Single __shared__ variable -> allocated at LDS offset 0, so the raw byte
    // offsets we hand to the async-load instruction line up with normal C++ reads.
    __shared__ v4f stage[2][256];           // 8 KB double buffer

    // Launder the LDS base pointer through an asm so the array ESCAPES: the
    // memory-clobbering async-load asms below must then be assumed to write it
    // (otherwise clang folds the uninitialized-shared read to undef and
    // constant-folds the entire kernel away, as seen in round 1).
    v4f (*sb)[256] = stage;
    asm volatile("" : "+v"(sb));

    const int tid     = threadIdx.x;
    const int tstride = gridDim.x;
    int t = blockIdx.x;

    // Prologue: issue async load of first tile into buffer 0.
    if (t < W_TILES) {
        unsigned voff = (unsigned)(t * 256 + tid) * 16u;   // byte offset into eps
        unsigned lds  = (unsigned)(tid * 16);              // LDS byte address (buf 0)
        asm volatile("global_load_async_to_lds_b128 %0, %1, %2"
                     :: "v"(lds), "v"(voff), "s"(eps) : "memory");
    }

    int buf = 0;
    for (; t < W_TILES; t += tstride) {
        int tn = t + tstride;
        if (tn < W_TILES) {
            // Make sure our previous ds_load of buf^1 has fully retired before the
            // async engine overwrites that buffer (async LDS writes are unordered
            // w.r.t. this wave's DS ops).
            asm volatile("s_wait_dscnt 0" ::: "memory");
            unsigned voff = (unsigned)(tn * 256 + tid) * 16u;
            unsigned lds  = (unsigned)(((buf ^ 1) * 4096) + tid * 16);
            asm volatile("global_load_async_to_lds_b128 %0, %1, %2"
                         :: "v"(lds), "v"(voff), "s"(eps) : "memory");
            // Current tile's load is the older of the two in flight.
            asm volatile("s_wait_asynccnt 1" ::: "memory");
        } else {
            asm volatile("s_wait_asynccnt 0" ::: "memory");
        }

        // Read our lane's staged 16B. Re-attach addrspace info so this lowers
        // to ds_load_b128 instead of a flat load through the laundered pointer.
        v4f* sp = &sb[buf][tid];
        __builtin_assume(__builtin_amdgcn_is_shared((const void*)sp));
        v4f e = *sp;

        unsigned v = (unsigned)t * 256u + (unsigned)tid;   // float4 index
        unsigned b = v / 75u;                              // batch element
        unsigned r = v - b * 75u;                          // vec-in-element
        unsigned p = r % 3u;                               // dim phase of lane .x

        float m0 = zm[3u * b + 0], m1 = zm[3u * b + 1], m2 = zm[3u * b + 2];
        float l0 = zlv[3u * b + 0], l1 = zlv[3u * b + 1], l2 = zlv[3u * b + 2];
        float s0 = __expf(0.5f * l0);
        float s1 = __expf(0.5f * l1);
        float s2 = __expf(0.5f * l2);

        // Rotate (mean,std) triple by phase p: component k uses d = (p+k)%3.
        float A0, A1, A2, S0, S1, S2;
        if (p == 0u)      { A0 = m0; A1 = m1; A2 = m2; S0 = s0; S1 = s1; S2 = s2; }
        else if (p == 1u) { A0 = m1; A1 = m2; A2 = m0; S0 = s1; S1 = s2; S2 = s0; }
        else              { A0 = m2; A1 = m0; A2 = m1; S0 = s2; S1 = s0; S2 = s1; }

        v4f res;
        res.x = fmaf(S0, e.x, A0);
        res.y = fmaf(S1, e.y, A1);
        res.z = fmaf(S2, e.z, A2);
        res.w = fmaf(S0, e.w, A0);   // (p+3)%3 == p

        __builtin_nontemporal_store(res, (v4f*)out + v);
        buf ^= 1;
    }
}

// labels_ext[b*100 + i] = (float)labels[b]; pure broadcast, b128 NT stores.
__global__ __launch_bounds__(256) void sampling_labels_kernel(
    const int* __restrict__ labels, float* __restrict__ outl)
{
    unsigned q      = blockIdx.x * 256u + threadIdx.x;
    unsigned stride = gridDim.x * 256u;
    for (; q < (unsigned)L_VECS; q += stride) {
        unsigned b = q / 25u;                 // 25 float4s per batch element
        float val  = (float)labels[b];
        v4f o = { val, val, val, val };
        __builtin_nontemporal_store(o, (v4f*)outl + q);
    }
}

extern "C" void kernel_launch(void* const* d_in, const int* in_sizes, int n_in,
                              void* d_out, int out_size, void* d_ws, size_t ws_size,
                              hipStream_t stream) {
    (void)in_sizes; (void)n_in; (void)out_size; (void)d_ws; (void)ws_size;
    const float* zm     = (const float*)d_in[0];   // z_mean    [B,3]
    const float* zlv    = (const float*)d_in[1];   // z_log_var [B,3]
    const int*   labels = (const int*)d_in[2];     // labels    [B]
    const float* eps    = (const float*)d_in[3];   // epsilon   [B,100,3]
    float* out = (float*)d_out;                    // [w (B*300) | labels_ext (B*100)]

    sampling_w_kernel<<<3072, 256, 0, stream>>>(zm, zlv, eps, out);
    sampling_labels_kernel<<<3072, 256, 0, stream>>>(labels, out + (size_t)W_FLOATS);
}